// BertCrf_66090956751317
// MI455X (gfx1250) — compile-verified
//
#include <hip/hip_runtime.h>
#include <hip/hip_bf16.h>

typedef __attribute__((ext_vector_type(2))) float v2f;
typedef __attribute__((ext_vector_type(8))) float v8f;

#define BB 64
#define SS 512
#define HH 768
#define TT 9
#define KSTEPS (HH / 4)          // 192 K-steps of 4
#define NTILES ((BB * SS) / 16)  // 2048 M-tiles

// ---------------------------------------------------------------------------
// Kernel 1: emissions[b,s,t] = hidden_states[b,s,:] . fc_w[t,:] + fc_b[t]
// fp32 WMMA (V_WMMA_F32_16X16X4_F32), B staged in LDS in WMMA layout.
// ---------------------------------------------------------------------------
__global__ __launch_bounds__(256) void emissions_wmma_kernel(
    const float* __restrict__ hs, const float* __restrict__ fc_w,
    const float* __restrict__ fc_b, float* __restrict__ emis) {
  __shared__ v2f bTile[KSTEPS * 32];  // 48 KB: B = fc_w^T tile, N padded 9->16

  const int tid  = threadIdx.x;
  const int lane = tid & 31;
  const int wave = tid >> 5;

  // Stage B in the 32-bit 4x16 WMMA B layout:
  //   VGPR0: lanes 0-15 -> K=k0+0, lanes 16-31 -> K=k0+2
  //   VGPR1: lanes 0-15 -> K=k0+1, lanes 16-31 -> K=k0+3
  for (int slot = tid; slot < KSTEPS * 32; slot += 256) {
    const int s  = slot >> 5;
    const int ln = slot & 31;
    const int n  = ln & 15;
    const int ko = (ln < 16) ? 0 : 2;
    const int k0 = s * 4 + ko;
    v2f v;
    v.x = 0.0f;
    v.y = 0.0f;
    if (n < TT) {
      v.x = fc_w[n * HH + k0];
      v.y = fc_w[n * HH + k0 + 1];
    }
    bTile[slot] = v;
  }
  __syncthreads();

  const int mtile = blockIdx.x * 8 + wave;            // 0..2047
  const int mrow  = mtile * 16 + (lane & 15);         // A row (flattened b*S+s)
  const int koff  = (lane < 16) ? 0 : 2;              // A-layout K offset
  const float* __restrict__ arow = hs + (size_t)mrow * HH + koff;

  v8f acc = {};
#pragma unroll 4
  for (int s = 0; s < KSTEPS; ++s) {
    v2f a = *(const v2f*)(arow + s * 4);              // 8B-aligned global_load_b64
    v2f b = bTile[s * 32 + lane];                     // ds_load_b64
    acc = __builtin_amdgcn_wmma_f32_16x16x4_f32(
        /*neg_a=*/false, a, /*neg_b=*/false, b,
        /*c_mod=*/(short)0, acc, /*reuse_a=*/false, /*reuse_b=*/false);
  }

  // C/D layout: VGPR r -> M = r (lanes 0-15) or r+8 (lanes 16-31), N = lane&15
  const int n     = lane & 15;
  const int rbase = (lane < 16) ? 0 : 8;
  if (n < TT) {
    const float bias = fc_b[n];
#pragma unroll
    for (int r = 0; r < 8; ++r) {
      emis[(size_t)(mtile * 16 + rbase + r) * TT + n] = acc[r] + bias;
    }
  }
}

// ---------------------------------------------------------------------------
// Kernel 2: per-batch CRF log-likelihood. One wave32 per batch.
// Lanes 0..8 each own next-tag j in the forward recurrence; alpha broadcast
// across lanes with __shfl each of the 511 sequential steps.
// ---------------------------------------------------------------------------
__global__ __launch_bounds__(32) void crf_kernel(
    const float* __restrict__ emis, const int* __restrict__ amask,
    const int* __restrict__ labels, const float* __restrict__ start_t,
    const float* __restrict__ end_t, const float* __restrict__ trans,
    float* __restrict__ outpb) {
  const int b    = blockIdx.x;
  const int lane = threadIdx.x;
  const float* __restrict__ E  = emis + (size_t)b * SS * TT;
  const int* __restrict__ lab  = labels + b * SS;
  const int* __restrict__ mk   = amask + b * SS;

  // ---- numerator partial over s in [1, S) + mask count (lane-strided) ----
  float numpart = 0.0f;
  int   mcount  = 0;
  for (int s = lane; s < SS; s += 32) {
    const int m = mk[s];
    mcount += m;
    if (s >= 1) {
      const int lp = lab[s - 1];
      const int lc = lab[s];
      numpart += (float)m * (trans[lp * TT + lc] + E[(size_t)s * TT + lc]);
    }
  }
#pragma unroll
  for (int off = 16; off; off >>= 1) {
    numpart += __shfl_xor(numpart, off, 32);
    mcount  += __shfl_xor(mcount, off, 32);
  }

  // ---- denominator: forward algorithm, lanes 0..8 = tag j ----
  const int  j    = lane;
  const bool actj = (j < TT);
  float alpha = actj ? (start_t[j] + E[j]) : -1.0e30f;
  float tj[TT];
#pragma unroll
  for (int i = 0; i < TT; ++i) tj[i] = actj ? trans[i * TT + j] : 0.0f;

  for (int s = 1; s < SS; ++s) {
    float ai[TT];
    float mmax = -1.0e30f;
#pragma unroll
    for (int i = 0; i < TT; ++i) {
      const float av = __shfl(alpha, i, 32);  // broadcast alpha[i] from lane i
      ai[i] = av + tj[i];
      mmax  = fmaxf(mmax, ai[i]);
    }
    float ssum = 0.0f;
#pragma unroll
    for (int i = 0; i < TT; ++i) ssum += __expf(ai[i] - mmax);
    const float e   = actj ? E[(size_t)s * TT + j] : 0.0f;
    const float nxt = e + mmax + __logf(ssum);
    if (mk[s] != 0) alpha = nxt;  // alpha = where(mask, next, alpha)
  }

  // den = logsumexp_j(alpha[j] + end_transitions[j])
  const float v = actj ? (alpha + end_t[j]) : -1.0e30f;
  float mx = v;
#pragma unroll
  for (int off = 16; off; off >>= 1) mx = fmaxf(mx, __shfl_xor(mx, off, 32));
  float ex = __expf(v - mx);
#pragma unroll
  for (int off = 16; off; off >>= 1) ex += __shfl_xor(ex, off, 32);
  const float den = mx + __logf(ex);

  if (lane == 0) {
    const int   l0   = lab[0];
    const int   last = mcount - 1;
    const float num  = start_t[l0] + E[l0] + numpart + end_t[lab[last]];
    outpb[b] = num - den;
  }
}

// ---------------------------------------------------------------------------
// Kernel 3: deterministic reduction of 64 per-batch values -> scalar loss.
// ---------------------------------------------------------------------------
__global__ __launch_bounds__(32) void finalize_kernel(
    const float* __restrict__ pb, float* __restrict__ out) {
  const int lane = threadIdx.x;
  float s = pb[lane] + pb[lane + 32];
#pragma unroll
  for (int off = 16; off; off >>= 1) s += __shfl_xor(s, off, 32);
  if (lane == 0) out[0] = -s;
}

extern "C" void kernel_launch(void* const* d_in, const int* in_sizes, int n_in,
                              void* d_out, int out_size, void* d_ws,
                              size_t ws_size, hipStream_t stream) {
  const float* hs      = (const float*)d_in[0];  // (64,512,768) f32
  const int*   amask   = (const int*)d_in[1];    // (64,512) i32
  const int*   labels  = (const int*)d_in[2];    // (64,512) i32
  const float* fc_w    = (const float*)d_in[3];  // (9,768) f32
  const float* fc_b    = (const float*)d_in[4];  // (9,) f32
  const float* start_t = (const float*)d_in[5];  // (9,) f32
  const float* end_t   = (const float*)d_in[6];  // (9,) f32
  const float* trans   = (const float*)d_in[7];  // (9,9) f32

  float* emis  = (float*)d_ws;                   // 64*512*9 f32 = 1.18 MB
  float* outpb = emis + (size_t)BB * SS * TT;    // 64 f32 partials

  emissions_wmma_kernel<<<NTILES / 8, 256, 0, stream>>>(hs, fc_w, fc_b, emis);
  crf_kernel<<<BB, 32, 0, stream>>>(emis, amask, labels, start_t, end_t, trans,
                                    outpb);
  finalize_kernel<<<1, 32, 0, stream>>>(outpb, (float*)d_out);
}